// FinalLayer_17454747090954
// MI455X (gfx1250) — compile-verified
//
#include <hip/hip_runtime.h>

typedef float v2f __attribute__((ext_vector_type(2)));
typedef float v8f __attribute__((ext_vector_type(8)));

#define BB      4096
#define NN      17
#define DIM     512
#define TWO_DIM 1024
#define OUTC    3
#define EPS_LN  1e-6f

// workspace layout (in floats)
#define WS_MOD  0                       // 4096*1024 = 4,194,304 floats (16 MB)
#define WS_SC   (BB * TWO_DIM)          // 4096*512  = 2,097,152 floats (8 MB)
#define WS_L    (WS_SC + BB * DIM)      // 289 floats

// ---------------------------------------------------------------------------
// Kernel 0a: sc = silu(c)   (element-wise over 4096*512)
// ---------------------------------------------------------------------------
__global__ void silu_kernel(const float* __restrict__ c, float* __restrict__ sc, int n) {
    int i = blockIdx.x * blockDim.x + threadIdx.x;
    if (i < n) {
        float v = c[i];
        sc[i] = v / (1.0f + __expf(-v));
    }
}

// ---------------------------------------------------------------------------
// Kernel 0b: normalized Laplacian  L = I - D^{-1/2} A D^{-1/2}   (17x17)
// ---------------------------------------------------------------------------
__global__ void laplacian_kernel(const float* __restrict__ adj, float* __restrict__ L) {
    __shared__ float dsh[NN];
    int t = threadIdx.x;
    if (t < NN) {
        float s = 0.0f;
        for (int m = 0; m < NN; ++m) s += adj[t * NN + m];
        dsh[t] = rsqrtf(s);
    }
    __syncthreads();
    if (t < NN * NN) {
        int n = t / NN, m = t % NN;
        float v = -dsh[n] * adj[t] * dsh[m];
        if (n == m) v += 1.0f;
        L[t] = v;
    }
}

// ---------------------------------------------------------------------------
// Kernel 1: mod = silu(c) @ W_mod + b_mod  via V_WMMA_F32_16X16X4_F32
// M = 4096 (256 tiles), N = 1024 (64 tiles), K = 512 (128 steps of 4).
// One wave per 16x16 output tile; 8 waves per block.
// A-frag (16x4 f32): lanes 0-15 hold rows M=0..15; half-wave selects K pair.
//   v0 = A[M=l][K + 2*half], v1 = A[M=l][K + 2*half + 1]
// B-frag (4x16 f32): mirror layout, lanes hold N columns.
// D (16x16 f32, 8 VGPRs): vgpr r -> M = r + 8*half, N = lane&15.
// ---------------------------------------------------------------------------
__global__ __launch_bounds__(256) void mod_gemm_wmma(
        const float* __restrict__ sc,     // [4096, 512]
        const float* __restrict__ Wm,     // [512, 1024]
        const float* __restrict__ bmod,   // [1024]
        float* __restrict__ mod)          // [4096, 1024]
{
    const int lane = threadIdx.x & 31;
    const int warp = threadIdx.x >> 5;
    const int tile = blockIdx.x * 8 + warp;     // 0..16383
    const int mt   = tile >> 6;                 // 0..255
    const int nt   = tile & 63;                 // 0..63
    const int m0   = mt * 16;
    const int n0   = nt * 16;
    const int half = lane >> 4;                 // 0 or 1
    const int l15  = lane & 15;

    const float* aptr = sc + (size_t)(m0 + l15) * DIM + 2 * half;   // + k each iter
    const float* bcol = Wm + n0 + l15;                              // column base

    v8f acc = {};
    for (int k = 0; k < DIM; k += 4) {
        v2f a = *(const v2f*)(aptr + k);        // A[m][k+2h], A[m][k+2h+1]
        int krow = k + 2 * half;
        v2f b;
        b.x = bcol[(size_t)krow * TWO_DIM];     // B[k+2h][n]
        b.y = bcol[(size_t)(krow + 1) * TWO_DIM];
        acc = __builtin_amdgcn_wmma_f32_16x16x4_f32(
                /*neg_a=*/false, a, /*neg_b=*/false, b,
                /*c_mod=*/(short)0, acc, /*reuse_a=*/false, /*reuse_b=*/false);
    }

    float bm = bmod[n0 + l15];
#pragma unroll
    for (int r = 0; r < 8; ++r) {
        int m = m0 + r + 8 * half;
        mod[(size_t)m * TWO_DIM + n0 + l15] = acc[r] + bm;
    }
}

// ---------------------------------------------------------------------------
// Kernel 2: fused LayerNorm + modulate + Chebyshev (T0,T1,T2) + projection.
// One block per batch element. 544 threads = 17 waves, one wave per node n.
// LDS: xs[17][512] (reused x -> xm -> T1) + L + weights  ~= 54 KB.
// Identity used:  out = T0*(Wg0 - Wg2) + T1*Wg1 + (2*L*T1)*Wg2
//                 (since T2 = 2*L*T1 - T0), so T0 stays lane-private.
// ---------------------------------------------------------------------------
__global__ __launch_bounds__(544) void fused_cheb_kernel(
        const float* __restrict__ x,     // [4096, 17, 512]
        const float* __restrict__ mod,   // [4096, 1024] (shift | scale)
        const float* __restrict__ Lg,    // [17, 17]
        const float* __restrict__ Wg,    // [3, 512, 3]
        const float* __restrict__ bg,    // [3]
        float* __restrict__ out)         // [4096, 17, 3]
{
    __shared__ float xs[NN * DIM];              // 34816 B
    __shared__ float Ls[NN * NN];               // 1156 B
    __shared__ float WgU[DIM * OUTC];           // Wg0 - Wg2   (6144 B)
    __shared__ float Wg1s[DIM * OUTC];          // 6144 B
    __shared__ float Wg2s[DIM * OUTC];          // 6144 B

    const int tid  = threadIdx.x;
    const int lane = tid & 31;
    const int w    = tid >> 5;                  // node/row id 0..16
    const int b    = blockIdx.x;

    // ---- cooperative load of x[b] (2176 float4 = 544*4) ----
    const float4* xg  = (const float4*)(x + (size_t)b * NN * DIM);
    float4*       xs4 = (float4*)xs;
#pragma unroll
    for (int j = 0; j < 4; ++j) xs4[tid + j * 544] = xg[tid + j * 544];

    // ---- stage weights + L in LDS ----
    for (int idx = tid; idx < DIM * OUTC; idx += 544) {
        float w0 = Wg[idx];
        float w1 = Wg[DIM * OUTC + idx];
        float w2 = Wg[2 * DIM * OUTC + idx];
        WgU[idx]  = w0 - w2;
        Wg1s[idx] = w1;
        Wg2s[idx] = w2;
    }
    if (tid < NN * NN) Ls[tid] = Lg[tid];
    __syncthreads();

    // ---- per-row LayerNorm stats (wave w owns row w) ----
    float s = 0.0f, sq = 0.0f;
    float xv[16];
#pragma unroll
    for (int i = 0; i < 16; ++i) {
        float v = xs[w * DIM + lane + 32 * i];
        xv[i] = v; s += v; sq += v * v;
    }
#pragma unroll
    for (int off = 16; off; off >>= 1) {
        s  += __shfl_xor(s,  off, 32);
        sq += __shfl_xor(sq, off, 32);
    }
    const float mu   = s  * (1.0f / DIM);
    const float var  = sq * (1.0f / DIM) - mu * mu;
    const float rstd = rsqrtf(var + EPS_LN);

    // ---- modulate: T0 in registers, also written back to xs (own row only) ----
    const float* shp = mod + (size_t)b * TWO_DIM;
    float t0[16];
#pragma unroll
    for (int i = 0; i < 16; ++i) {
        int c = lane + 32 * i;
        float sh  = shp[c];
        float scl = shp[DIM + c];
        float v = (xv[i] - mu) * rstd;
        t0[i] = v * (1.0f + scl) + sh;
        xs[w * DIM + c] = t0[i];
    }
    __syncthreads();

    // ---- T1 = L * xm (registers), then swap into same LDS buffer ----
    float Lrow[NN];
#pragma unroll
    for (int m = 0; m < NN; ++m) Lrow[m] = Ls[w * NN + m];

    float t1[16];
#pragma unroll
    for (int i = 0; i < 16; ++i) {
        int c = lane + 32 * i;
        float a = 0.0f;
#pragma unroll
        for (int m = 0; m < NN; ++m) a += Lrow[m] * xs[m * DIM + c];
        t1[i] = a;
    }
    __syncthreads();                     // everyone done reading xm
#pragma unroll
    for (int i = 0; i < 16; ++i) xs[w * DIM + lane + 32 * i] = t1[i];
    __syncthreads();                     // T1 visible to all waves

    // ---- projection: out = t0*(Wg0-Wg2) + t1*Wg1 + (2*L*T1)*Wg2 ----
    float p0 = 0.0f, p1 = 0.0f, p2 = 0.0f;
#pragma unroll
    for (int i = 0; i < 16; ++i) {
        int c = lane + 32 * i;
        float t2 = 0.0f;
#pragma unroll
        for (int m = 0; m < NN; ++m) t2 += Lrow[m] * xs[m * DIM + c];
        t2 *= 2.0f;
        float a0 = t0[i], a1 = t1[i];
        p0 += a0 * WgU[c * 3 + 0] + a1 * Wg1s[c * 3 + 0] + t2 * Wg2s[c * 3 + 0];
        p1 += a0 * WgU[c * 3 + 1] + a1 * Wg1s[c * 3 + 1] + t2 * Wg2s[c * 3 + 1];
        p2 += a0 * WgU[c * 3 + 2] + a1 * Wg1s[c * 3 + 2] + t2 * Wg2s[c * 3 + 2];
    }
#pragma unroll
    for (int off = 16; off; off >>= 1) {
        p0 += __shfl_xor(p0, off, 32);
        p1 += __shfl_xor(p1, off, 32);
        p2 += __shfl_xor(p2, off, 32);
    }
    if (lane == 0) {
        float* o = out + (size_t)b * NN * OUTC + w * OUTC;
        o[0] = p0 + bg[0];
        o[1] = p1 + bg[1];
        o[2] = p2 + bg[2];
    }
}

// ---------------------------------------------------------------------------
// Host launcher
// ---------------------------------------------------------------------------
extern "C" void kernel_launch(void* const* d_in, const int* in_sizes, int n_in,
                              void* d_out, int out_size, void* d_ws, size_t ws_size,
                              hipStream_t stream) {
    const float* x     = (const float*)d_in[0];   // [4096,17,512]
    const float* adj   = (const float*)d_in[1];   // [17,17]
    const float* c     = (const float*)d_in[2];   // [4096,1,512]
    const float* W_mod = (const float*)d_in[3];   // [512,1024]
    const float* b_mod = (const float*)d_in[4];   // [1024]
    const float* Wg    = (const float*)d_in[5];   // [3,512,3]
    const float* bg    = (const float*)d_in[6];   // [3]
    float*       out   = (float*)d_out;           // [4096,17,3]

    float* wsf = (float*)d_ws;
    float* mod = wsf + WS_MOD;                    // [4096,1024]
    float* sc  = wsf + WS_SC;                     // [4096,512]
    float* Lws = wsf + WS_L;                      // [289]

    // 0a: silu(c)
    const int nsilu = BB * DIM;
    silu_kernel<<<(nsilu + 255) / 256, 256, 0, stream>>>(c, sc, nsilu);

    // 0b: normalized Laplacian
    laplacian_kernel<<<1, 320, 0, stream>>>(adj, Lws);

    // 1: WMMA modulation GEMM -> mod
    // tiles = (4096/16) * (1024/16) = 16384; 8 waves (tiles) per block
    mod_gemm_wmma<<<16384 / 8, 256, 0, stream>>>(sc, W_mod, b_mod, mod);

    // 2: fused LayerNorm + modulate + Chebyshev + projection
    fused_cheb_kernel<<<BB, 544, 0, stream>>>(x, mod, Lws, Wg, bg, out);
}